// GCN_962072674854
// MI455X (gfx1250) — compile-verified
//
#include <hip/hip_runtime.h>
#include <hip/hip_bf16.h>
#include <math.h>

typedef float v2f __attribute__((ext_vector_type(2)));
typedef float v8f __attribute__((ext_vector_type(8)));

#define DIN  512
#define DH   128
#define DOUT 40
#define DOP  48   // DOUT padded to 3 WMMA tiles

__device__ __forceinline__ v8f wmma4(v2f a, v2f b, v8f c) {
  return __builtin_amdgcn_wmma_f32_16x16x4_f32(false, a, false, b, (short)0, c, false, false);
}

// ---------------- graph preprocessing ----------------
__global__ void k_init(float* __restrict__ dinv, int* __restrict__ cnt,
                       int* __restrict__ cursor, int n) {
  int t = blockIdx.x * blockDim.x + threadIdx.x;
  if (t < n) { dinv[t] = 0.0f; cnt[t] = 0; cursor[t] = 0; }
}

__global__ void k_count(const int* __restrict__ dst, int* __restrict__ cnt, int E) {
  int t = blockIdx.x * blockDim.x + threadIdx.x;
  if (t < E) atomicAdd(&cnt[dst[t]], 1);
}

__global__ void k_dinv(const int* __restrict__ cnt, float* __restrict__ dinv, int n) {
  int t = blockIdx.x * blockDim.x + threadIdx.x;
  if (t < n) dinv[t] = rsqrtf((float)cnt[t] + 1.0f);  // +1 = self-loop
}

__global__ void k_norm(const int* __restrict__ src, const int* __restrict__ dst,
                       const float* __restrict__ dinv, float* __restrict__ norm, int E) {
  int t = blockIdx.x * blockDim.x + threadIdx.x;
  if (t < E) norm[t] = dinv[src[t]] * dinv[dst[t]];
}

// exclusive scan of cnt -> rowstart (block-local), partials[blk] = block sum
__global__ void k_scan1(const int* __restrict__ cnt, int* __restrict__ rowstart,
                        int* __restrict__ partials, int n) {
  __shared__ int sh[256];
  int t = blockIdx.x * 256 + threadIdx.x;
  int v = (t < n) ? cnt[t] : 0;
  sh[threadIdx.x] = v;
  __syncthreads();
  for (int off = 1; off < 256; off <<= 1) {
    int x = (threadIdx.x >= off) ? sh[threadIdx.x - off] : 0;
    __syncthreads();
    sh[threadIdx.x] += x;
    __syncthreads();
  }
  if (t < n) rowstart[t] = sh[threadIdx.x] - v;
  if (threadIdx.x == 255) partials[blockIdx.x] = sh[255];
}

// single-block exclusive scan of partials (P <= 512)
__global__ void k_scan2(int* __restrict__ partials, int P) {
  __shared__ int sh[512];
  int t = threadIdx.x;
  int v = (t < P) ? partials[t] : 0;
  sh[t] = v;
  __syncthreads();
  for (int off = 1; off < 512; off <<= 1) {
    int x = (t >= off) ? sh[t - off] : 0;
    __syncthreads();
    sh[t] += x;
    __syncthreads();
  }
  if (t < P) partials[t] = sh[t] - v;
}

__global__ void k_scan3(int* __restrict__ rowstart, const int* __restrict__ partials, int n) {
  int t = blockIdx.x * blockDim.x + threadIdx.x;
  if (t < n) rowstart[t] += partials[t >> 8];
}

// place edge ids into CSR slots (cheap int atomics; order fixed up by sort below)
__global__ void k_place(const int* __restrict__ dst, const int* __restrict__ rowstart,
                        int* __restrict__ cursor, int* __restrict__ eids, int E) {
  int t = blockIdx.x * blockDim.x + threadIdx.x;
  if (t < E) {
    int d = dst[t];
    int pos = rowstart[d] + atomicAdd(&cursor[d], 1);
    eids[pos] = t;
  }
}

// insertion-sort each node's edge-id segment -> deterministic summation order
__global__ void k_sortseg(int* __restrict__ eids, const int* __restrict__ rowstart,
                          const int* __restrict__ cnt, int n) {
  int i = blockIdx.x * blockDim.x + threadIdx.x;
  if (i >= n) return;
  int s0 = rowstart[i], c = cnt[i];
  for (int a = 1; a < c; ++a) {
    int key = eids[s0 + a];
    int b = a - 1;
    while (b >= 0 && eids[s0 + b] > key) { eids[s0 + b + 1] = eids[s0 + b]; --b; }
    eids[s0 + b + 1] = key;
  }
}

__global__ void k_padW2(const float* __restrict__ W2, float* __restrict__ W2p) {
  int t = blockIdx.x * blockDim.x + threadIdx.x;
  if (t < DH * DOP) {
    int r = t / DOP, c = t % DOP;
    W2p[t] = (c < DOUT) ? W2[r * DOUT + c] : 0.0f;
  }
}

// ---------------- GEMM1: h1 = x @ W1  (M x 512 @ 512 x 128) ----------------
// 2x2 register blocking (4 acc tiles/wave, 1 LDS fragment per WMMA).
// Latency hiding comes from multi-block residency (49 KB LDS of 320 KB/WGP ->
// ~6 blocks per WGP) plus global_prefetch_b8 of the next K-chunk; no register
// staging, so no scratch spills.
__global__ void __launch_bounds__(256) k_gemm1(const float* __restrict__ A,
                                               const float* __restrict__ B,
                                               float* __restrict__ C, int M) {
  __shared__ float As[64 * 68];   // stride 68: conflict-free A-frag reads
  __shared__ float Bs[64 * DH];
  const int tid  = threadIdx.x;
  const int wave = tid >> 5;
  const int lane = tid & 31;
  const int lm = lane & 15;
  const int lh = lane >> 4;
  const int wm = wave >> 2;           // 0..1  -> M-tiles {2wm, 2wm+1}
  const int wn = wave & 3;            // 0..3  -> N-tiles {2wn, 2wn+1}
  const int m0 = wm * 32;
  const int n0 = wn * 32;
  const int row0 = blockIdx.x * 64;

  // per-thread staging indices (fixed across chunks)
  const int ar  = tid >> 4;
  const int ac4 = (tid & 15) << 2;
  const int br  = tid >> 5;
  const int bc4 = (tid & 31) << 2;
  int arow[4];
#pragma unroll
  for (int i = 0; i < 4; ++i) {       // clamp A rows once
    int rr = row0 + ar + (i << 4);
    arow[i] = (rr >= M) ? (M - 1) : rr;
  }

  v8f a00 = {}, a01 = {}, a10 = {}, a11 = {};
  for (int kc = 0; kc < DIN; kc += 64) {
    __syncthreads();                  // previous chunk's compute done
#pragma unroll
    for (int i = 0; i < 4; ++i)
      *(float4*)(&As[(ar + (i << 4)) * 68 + ac4]) =
          *(const float4*)(A + (size_t)arow[i] * DIN + kc + ac4);
#pragma unroll
    for (int i = 0; i < 8; ++i)
      *(float4*)(&Bs[(br + (i << 3)) * DH + bc4]) =
          *(const float4*)(B + (size_t)(kc + br + (i << 3)) * DH + bc4);
    __syncthreads();
    if (kc + 64 < DIN) {              // warm L2/L0 for the next chunk
      const int kn = kc + 64;
#pragma unroll
      for (int i = 0; i < 4; ++i)
        __builtin_prefetch(A + (size_t)arow[i] * DIN + kn + ac4, 0, 0);
#pragma unroll
      for (int i = 0; i < 8; ++i)
        __builtin_prefetch(B + (size_t)(kn + br + (i << 3)) * DH + bc4, 0, 0);
    }
#pragma unroll
    for (int kk = 0; kk < 64; kk += 4) {
      const int ka = kk + lh * 2;     // A frag: M=lm, K = ka+{0,1}
      v2f f0, f1, g0, g1;
      f0.x = As[(m0 + lm) * 68 + ka];      f0.y = As[(m0 + lm) * 68 + ka + 1];
      f1.x = As[(m0 + 16 + lm) * 68 + ka]; f1.y = As[(m0 + 16 + lm) * 68 + ka + 1];
      g0.x = Bs[ka * DH + n0 + lm];        g0.y = Bs[(ka + 1) * DH + n0 + lm];
      g1.x = Bs[ka * DH + n0 + 16 + lm];   g1.y = Bs[(ka + 1) * DH + n0 + 16 + lm];
      a00 = wmma4(f0, g0, a00);
      a01 = wmma4(f0, g1, a01);
      a10 = wmma4(f1, g0, a10);
      a11 = wmma4(f1, g1, a11);
    }
  }
#pragma unroll
  for (int v = 0; v < 8; ++v) {       // C/D layout: row = v + 8*lh, col = lm
    int r0 = row0 + m0 + v + 8 * lh;
    if (r0 < M) {
      C[(size_t)r0 * DH + n0 + lm]      = a00[v];
      C[(size_t)r0 * DH + n0 + 16 + lm] = a01[v];
    }
    if (r0 + 16 < M) {
      C[(size_t)(r0 + 16) * DH + n0 + lm]      = a10[v];
      C[(size_t)(r0 + 16) * DH + n0 + 16 + lm] = a11[v];
    }
  }
}

// ---------------- GEMM2: h2 = h1r @ W2p  (M x 128 @ 128 x 48) ----------------
// 8 M-tiles per block: W2 staged in LDS once per 128 output rows.
#define G2_TILES 8
__global__ void __launch_bounds__(96) k_gemm2(const float* __restrict__ A,
                                              const float* __restrict__ B,
                                              float* __restrict__ C, int M) {
  __shared__ float As[16 * 132];
  __shared__ float Bs[DH * DOP];
  const int tid  = threadIdx.x;   // 96 threads = 3 waves
  const int wave = tid >> 5;
  const int lane = tid & 31;
  const int lm = lane & 15;
  const int lh = lane >> 4;
  const int n0 = wave * 16;
  for (int idx = tid; idx < 1536; idx += 96) {       // whole padded W2 once
    int r  = idx / 12;
    int c4 = (idx % 12) << 2;
    *(float4*)(&Bs[r * DOP + c4]) = *(const float4*)(B + (size_t)r * DOP + c4);
  }
  const int base = blockIdx.x * (16 * G2_TILES);
  for (int tm = 0; tm < G2_TILES; ++tm) {
    const int row0 = base + tm * 16;                 // uniform across block
    if (row0 >= M) break;
    __syncthreads();                                 // As reuse + first-iter Bs
    for (int idx = tid; idx < 512; idx += 96) {
      int r = idx >> 5;
      int c4 = (idx & 31) << 2;
      int rr = row0 + r; if (rr >= M) rr = M - 1;
      *(float4*)(&As[r * 132 + c4]) = *(const float4*)(A + (size_t)rr * DH + c4);
    }
    __syncthreads();
    v8f acc = {};
#pragma unroll
    for (int kk = 0; kk < DH; kk += 4) {
      v2f a, b;
      a.x = As[lm * 132 + kk + lh * 2 + 0];
      a.y = As[lm * 132 + kk + lh * 2 + 1];
      b.x = Bs[(kk + lh * 2 + 0) * DOP + n0 + lm];
      b.y = Bs[(kk + lh * 2 + 1) * DOP + n0 + lm];
      acc = wmma4(a, b, acc);
    }
#pragma unroll
    for (int v = 0; v < 8; ++v) {
      int r0 = row0 + v + 8 * lh;
      if (r0 < M) C[(size_t)r0 * DOP + n0 + lm] = acc[v];
    }
  }
}

// ---------------- layer-1 gather: CSR segment sum + self-loop + bias + ReLU ----------
// one wave per node; 32 lanes x float4 = all 128 features; next-edge row prefetch
__global__ void k_gather1(const float* __restrict__ h1, const int* __restrict__ src,
                          const float* __restrict__ norm, const int* __restrict__ eids,
                          const int* __restrict__ rowstart, const int* __restrict__ cnt,
                          const float* __restrict__ dinv, const float* __restrict__ b1,
                          float* __restrict__ out, int N) {
  int node = blockIdx.x * (blockDim.x >> 5) + (threadIdx.x >> 5);
  int lane = threadIdx.x & 31;
  if (node >= N) return;                      // uniform per wave
  const int f4 = lane << 2;
  float di = dinv[node];
  float sl = di * di;
  float4 acc = *(const float4*)(h1 + (size_t)node * DH + f4);
  acc.x *= sl; acc.y *= sl; acc.z *= sl; acc.w *= sl;
  int s0 = rowstart[node], c = cnt[node];
  for (int j = 0; j < c; ++j) {
    int e = eids[s0 + j];                     // sorted -> deterministic order
    int s = src[e];
    float nm = norm[e];
    if (j + 1 < c) {                          // lookahead: global_prefetch_b8
      int s2 = src[eids[s0 + j + 1]];
      __builtin_prefetch(h1 + (size_t)s2 * DH + f4, 0, 0);
    }
    const float4 v = *(const float4*)(h1 + (size_t)s * DH + f4);
    acc.x += v.x * nm; acc.y += v.y * nm; acc.z += v.z * nm; acc.w += v.w * nm;
  }
  const float4 bb = *(const float4*)(b1 + f4);
  float4 r;
  r.x = fmaxf(acc.x + bb.x, 0.0f);
  r.y = fmaxf(acc.y + bb.y, 0.0f);
  r.z = fmaxf(acc.z + bb.z, 0.0f);
  r.w = fmaxf(acc.w + bb.w, 0.0f);
  *(float4*)(out + (size_t)node * DH + f4) = r;
}

// ---------------- layer-2 gather fused with bias + log_softmax ----------------
// one wave per node; lane holds f=lane, and f=32+lane for lane<8
__global__ void k_gather2(const float* __restrict__ h2, const int* __restrict__ src,
                          const float* __restrict__ norm, const int* __restrict__ eids,
                          const int* __restrict__ rowstart, const int* __restrict__ cnt,
                          const float* __restrict__ dinv, const float* __restrict__ b2,
                          float* __restrict__ out, int N) {
  int node = blockIdx.x * (blockDim.x >> 5) + (threadIdx.x >> 5);
  int lane = threadIdx.x & 31;
  if (node >= N) return;
  float di = dinv[node];
  float sl = di * di;
  const float* hrow = h2 + (size_t)node * DOP;
  float a = hrow[lane] * sl;
  float b = (lane < 8) ? hrow[32 + lane] * sl : 0.0f;
  int s0 = rowstart[node], c = cnt[node];
  for (int j = 0; j < c; ++j) {
    int e = eids[s0 + j];
    int s = src[e];
    float nm = norm[e];
    if (j + 1 < c) {
      int s2 = src[eids[s0 + j + 1]];
      __builtin_prefetch(h2 + (size_t)s2 * DOP + lane, 0, 0);
    }
    const float* r = h2 + (size_t)s * DOP;
    a += r[lane] * nm;
    if (lane < 8) b += r[32 + lane] * nm;
  }
  a += b2[lane];
  b = (lane < 8) ? (b + b2[32 + lane]) : -INFINITY;
  float m = fmaxf(a, b);
#pragma unroll
  for (int off = 16; off > 0; off >>= 1) m = fmaxf(m, __shfl_xor(m, off, 32));
  float s = expf(a - m) + ((lane < 8) ? expf(b - m) : 0.0f);
#pragma unroll
  for (int off = 16; off > 0; off >>= 1) s += __shfl_xor(s, off, 32);
  float ls = logf(s);
  out[(size_t)node * DOUT + lane] = a - m - ls;
  if (lane < 8) out[(size_t)node * DOUT + 32 + lane] = b - m - ls;
}

// ---------------- launcher ----------------
extern "C" void kernel_launch(void* const* d_in, const int* in_sizes, int n_in,
                              void* d_out, int out_size, void* d_ws, size_t ws_size,
                              hipStream_t stream) {
  const float* x  = (const float*)d_in[0];
  const int*   ei = (const int*)d_in[1];      // jnp defaults to int32 (x64 off)
  const float* W1 = (const float*)d_in[2];
  const float* b1 = (const float*)d_in[3];
  const float* W2 = (const float*)d_in[4];
  const float* b2 = (const float*)d_in[5];
  const int N = in_sizes[0] / DIN;            // 100000
  const int E = in_sizes[1] / 2;              // 1600000
  const int* src = ei;
  const int* dst = ei + E;

  // workspace carve-up (4-byte elems, all offsets 16B-aligned)
  float* ws      = (float*)d_ws;
  float* dinv    = ws;                         // N
  float* norm    = dinv + N;                   // E
  float* h1      = norm + E;                   // N*128
  float* h1r     = h1 + (size_t)N * DH;        // N*128
  float* h2      = h1r + (size_t)N * DH;       // N*48
  float* W2p     = h2 + (size_t)N * DOP;       // 128*48
  int* cnt       = (int*)(W2p + DH * DOP);     // N
  int* cursor    = cnt + N;                    // N
  int* rowstart  = cursor + N;                 // N
  int* partials  = rowstart + N;               // 512
  int* eids      = partials + 512;             // E

  const int P = (N + 255) / 256;               // 391 partial blocks (<=512)

  // degrees, normalization, CSR build (deterministic after per-segment sort)
  k_init  <<<(N + 255) / 256, 256, 0, stream>>>(dinv, cnt, cursor, N);
  k_count <<<(E + 255) / 256, 256, 0, stream>>>(dst, cnt, E);
  k_dinv  <<<(N + 255) / 256, 256, 0, stream>>>(cnt, dinv, N);
  k_norm  <<<(E + 255) / 256, 256, 0, stream>>>(src, dst, dinv, norm, E);
  k_scan1 <<<P, 256, 0, stream>>>(cnt, rowstart, partials, N);
  k_scan2 <<<1, 512, 0, stream>>>(partials, P);
  k_scan3 <<<(N + 255) / 256, 256, 0, stream>>>(rowstart, partials, N);
  k_place <<<(E + 255) / 256, 256, 0, stream>>>(dst, rowstart, cursor, eids, E);
  k_sortseg <<<(N + 255) / 256, 256, 0, stream>>>(eids, rowstart, cnt, N);

  // layer 1: WMMA GEMM + CSR gather (self-loop + bias + ReLU fused)
  k_gemm1 <<<(N + 63) / 64, 256, 0, stream>>>(x, W1, h1, N);
  k_gather1 <<<(N + 7) / 8, 256, 0, stream>>>(h1, src, norm, eids, rowstart, cnt,
                                              dinv, b1, h1r, N);

  // layer 2: WMMA GEMM (W2 staged once per 128 rows) + gather/bias/log_softmax
  k_padW2 <<<(DH * DOP + 255) / 256, 256, 0, stream>>>(W2, W2p);
  k_gemm2 <<<(N + 16 * G2_TILES - 1) / (16 * G2_TILES), 96, 0, stream>>>(h1r, W2p, h2, N);
  k_gather2 <<<(N + 7) / 8, 256, 0, stream>>>(h2, src, norm, eids, rowstart, cnt,
                                              dinv, b2, (float*)d_out, N);
}